// PointPillarScatter_79173427134756
// MI455X (gfx1250) — compile-verified
//
#include <hip/hip_runtime.h>

// PointPillarScatter for MI455X (gfx1250, wave32).
// Pure data movement (0 FLOPs): gather formulation, ~308 MB total traffic
// -> ~13 us at 23.3 TB/s HBM. Output stream (256 MB > 192 MB L2) uses
// non-temporal store hints (TH_STORE_NT).

#define BB    4
#define CC    64
#define NXX   512
#define NYY   512
#define PLANE (NYY * NXX)          // 262144 elements per (b,c) plane
#define MAPN  (BB * PLANE)         // 1,048,576 map entries

// Native clang vector type: __builtin_nontemporal_store requires a scalar or
// vector-of-scalar type, not HIP's int4 class.
typedef int   v4i __attribute__((ext_vector_type(4)));
typedef float v4f __attribute__((ext_vector_type(4)));

// ---------------------------------------------------------------------------
// Kernel 1: init pillar-index map to -1 (vectorized b128 stores, NT hint)
// ---------------------------------------------------------------------------
__global__ void pps_init_map(v4i* __restrict__ map4, int n4) {
    int i = blockIdx.x * blockDim.x + threadIdx.x;
    if (i < n4) {
        v4i m1 = {-1, -1, -1, -1};
        __builtin_nontemporal_store(m1, map4 + i);
    }
}

// ---------------------------------------------------------------------------
// Kernel 2: scatter pillar id into map. atomicMax makes duplicate-coordinate
// resolution deterministic (highest pillar id wins) -> GLOBAL_ATOMIC_MAX_I32.
// ---------------------------------------------------------------------------
__global__ void pps_build_map(const int* __restrict__ coords,
                              int* __restrict__ map, int P) {
    int p = blockIdx.x * blockDim.x + threadIdx.x;
    if (p < P) {
        int b = coords[4 * p + 0];
        int z = coords[4 * p + 1];   // always 0 (nz == 1), kept for fidelity
        int y = coords[4 * p + 2];
        int x = coords[4 * p + 3];
        int lin = b * PLANE + z + y * NXX + x;
        atomicMax(&map[lin], p);
    }
}

// ---------------------------------------------------------------------------
// Kernel 3: gather. One thread per (b, y, x). Lanes cover consecutive x, so
// each per-channel store is a fully coalesced 128B line (32 lanes x 4B).
// Feature loads are EXEC-predicated (only ~2% of threads hit a pillar) but
// the register staging keeps the 64 NT stores under full EXEC, so store
// coalescing is never split by divergence.
// ---------------------------------------------------------------------------
__global__ void __launch_bounds__(256)
pps_gather(const float* __restrict__ pf,       // [P, C]
           const int*   __restrict__ map,      // [B*NY*NX]
           float*       __restrict__ out) {    // [B, C, NY, NX]
    int idx = blockIdx.x * blockDim.x + threadIdx.x;   // b*PLANE + y*NX + x
    if (idx >= MAPN) return;

    int pid = __builtin_nontemporal_load(map + idx);
    int b   = idx >> 18;                 // PLANE = 1<<18, idx < 4*PLANE
    int pos = idx & (PLANE - 1);         // y*NX + x

    float* outp = out + (size_t)b * CC * PLANE + pos;

    // Stage 64 channel values in registers (zeros unless this cell has a pillar)
    v4f v[CC / 4];
#pragma unroll
    for (int i = 0; i < CC / 4; ++i) v[i] = (v4f){0.f, 0.f, 0.f, 0.f};

    if (pid >= 0) {
        const v4f* row = (const v4f*)(pf + (size_t)pid * CC);
        __builtin_prefetch(row, 0, 0);   // global_prefetch_b8 (gfx1250 path)
#pragma unroll
        for (int i = 0; i < CC / 4; ++i) v[i] = row[i];   // 16x b128 loads
    }

    // 64 coalesced non-temporal stores, one per channel plane, full EXEC
#pragma unroll
    for (int i = 0; i < CC / 4; ++i) {
        __builtin_nontemporal_store(v[i].x, outp + (size_t)(4 * i + 0) * PLANE);
        __builtin_nontemporal_store(v[i].y, outp + (size_t)(4 * i + 1) * PLANE);
        __builtin_nontemporal_store(v[i].z, outp + (size_t)(4 * i + 2) * PLANE);
        __builtin_nontemporal_store(v[i].w, outp + (size_t)(4 * i + 3) * PLANE);
    }
}

// ---------------------------------------------------------------------------
// Launch: init map -> scatter ids -> gather/write. All on `stream`,
// deterministic, no allocation, scratch (d_ws) re-initialized every call.
// ---------------------------------------------------------------------------
extern "C" void kernel_launch(void* const* d_in, const int* in_sizes, int n_in,
                              void* d_out, int out_size, void* d_ws, size_t ws_size,
                              hipStream_t stream) {
    const float* pillar_features = (const float*)d_in[0];   // [P, C] f32
    const int*   coords          = (const int*)d_in[1];     // [P, 4] i32
    // d_in[2] (voxel_features) is unused by the reference output.

    const int P = in_sizes[1] / 4;
    int* map = (int*)d_ws;    // needs MAPN * 4 = 4 MB of scratch

    // 1) map[:] = -1
    {
        int n4 = MAPN / 4;
        dim3 grid((n4 + 255) / 256), block(256);
        pps_init_map<<<grid, block, 0, stream>>>((v4i*)map, n4);
    }
    // 2) map[lin(p)] = max(p)
    {
        dim3 grid((P + 255) / 256), block(256);
        pps_build_map<<<grid, block, 0, stream>>>(coords, map, P);
    }
    // 3) out[b,c,y,x] = (map hit) ? pillar_features[pid, c] : 0
    {
        dim3 grid(MAPN / 256), block(256);
        pps_gather<<<grid, block, 0, stream>>>(pillar_features, map, (float*)d_out);
    }
}